// Attn_32615981646491
// MI455X (gfx1250) — compile-verified
//
#include <hip/hip_runtime.h>

typedef float v2f __attribute__((ext_vector_type(2)));
typedef float v4f __attribute__((ext_vector_type(4)));
typedef float v8f __attribute__((ext_vector_type(8)));

#define Hdim 1024   // hidden size
#define Bdim 32     // batch
#define Sdim 2048   // sequence length
#define K2   2048   // 2*H

// ---------------------------------------------------------------------------
// Kernel 1: v[b,k] = sum_h hidden[b,h] * W[h,k]   (32x2048 = 32x1024 @ 1024x2048)
// fp32 WMMA: D(16x16) += A(16x4) x B(4x16), one wave per output tile.
// 2 M-tiles x 128 N-tiles = 256 waves, K loop = 1024/4 = 256 wmma ops/wave.
// EXEC is all-ones (no divergence) as required for WMMA.
// ---------------------------------------------------------------------------
__global__ void __launch_bounds__(256)
proj_wmma(const float* __restrict__ hidden,
          const float* __restrict__ W,
          float* __restrict__ v) {
  const int lane = threadIdx.x & 31;
  const int warp = threadIdx.x >> 5;
  const int tile = blockIdx.x * 8 + warp;     // 0..255
  const int mrow = tile & 1;                  // which half of the 32 batches
  const int n0   = (tile >> 1) << 4;          // N tile base (0..2032)

  // A fragment addressing (32-bit A 16x4): lanes 0-15 -> K=0,1 ; lanes 16-31 -> K=2,3
  const int mA = mrow * 16 + (lane & 15);     // M row for this lane
  const int hi = (lane >= 16) ? 2 : 0;        // K sub-offset
  const int nB = n0 + (lane & 15);            // N column for B fragment

  v8f acc = {};
  for (int k0 = 0; k0 < Hdim; k0 += 4) {
    const int ka = k0 + hi;
    v2f a, bm;
    a.x  = hidden[mA * Hdim + ka];            // A[M, ka]
    a.y  = hidden[mA * Hdim + ka + 1];        // A[M, ka+1]
    bm.x = W[ka * K2 + nB];                   // B[ka,   n]
    bm.y = W[(ka + 1) * K2 + nB];             // B[ka+1, n]
    // 8 args: (neg_a, A, neg_b, B, c_mod, C, reuse_a, reuse_b)
    acc = __builtin_amdgcn_wmma_f32_16x16x4_f32(false, a, false, bm,
                                                (short)0, acc, false, false);
  }

  // D layout: VGPR r -> M = base+r (lanes 0-15) / base+8+r (lanes 16-31), N = n0+(lane&15)
  const int mbase = mrow * 16 + ((lane >= 16) ? 8 : 0);
  const int col   = n0 + (lane & 15);
  #pragma unroll
  for (int r = 0; r < 8; ++r)
    v[(mbase + r) * K2 + col] = acc[r];
}

// ---------------------------------------------------------------------------
// Kernel 2 (the 512 MB streaming pass — bandwidth bound, ~22us floor):
// scores[b,s] = v[b] . enc[s,b].  One wave32 per (s,b) row of 8 KB.
// enc is read once -> non-temporal b128 loads; v (256 KB) stays cache-hot.
// ---------------------------------------------------------------------------
__global__ void __launch_bounds__(256)
scores_kernel(const float* __restrict__ enc,
              const float* __restrict__ v,
              float* __restrict__ scores) {
  const int lane = threadIdx.x & 31;
  const int wave = (blockIdx.x << 3) + (threadIdx.x >> 5);  // 8 waves / block
  const int s = wave >> 5;                                  // wave / 32
  const int b = wave & 31;                                  // consecutive b -> contiguous 64KB per block

  const float* __restrict__ erow = enc + ((size_t)s * Bdim + b) * K2;
  const float* __restrict__ vrow = v + b * K2;

  float acc = 0.0f;
  #pragma unroll
  for (int j = 0; j < 16; ++j) {
    const int idx = (lane + (j << 5)) << 2;  // float index; wave covers 512B/step contiguously
    v4f e = __builtin_nontemporal_load((const v4f*)(erow + idx));  // TH=NT: stream, don't pollute
    v4f w = *(const v4f*)(vrow + idx);                             // RT: L2-resident, reused S times
    acc += e.x * w.x + e.y * w.y + e.z * w.z + e.w * w.w;
  }

  // wave32 reduction
  #pragma unroll
  for (int off = 16; off > 0; off >>= 1)
    acc += __shfl_xor(acc, off, 32);

  if (lane == 0) scores[b * Sdim + s] = acc;
}

// ---------------------------------------------------------------------------
// Kernel 3: softmax over s for each b (32 rows x 2048). Bias term omitted:
// it is constant per row and softmax is shift-invariant.
// ---------------------------------------------------------------------------
__global__ void __launch_bounds__(256)
softmax_kernel(const float* __restrict__ scores, float* __restrict__ out) {
  __shared__ float red[256];
  const int b = blockIdx.x;
  const int t = threadIdx.x;
  const float* __restrict__ row = scores + b * Sdim;

  float vals[8];
  float lmax = -__builtin_inff();
  #pragma unroll
  for (int j = 0; j < 8; ++j) {
    vals[j] = row[t + j * 256];
    lmax = fmaxf(lmax, vals[j]);
  }
  red[t] = lmax; __syncthreads();
  for (int off = 128; off > 0; off >>= 1) {
    if (t < off) red[t] = fmaxf(red[t], red[t + off]);
    __syncthreads();
  }
  const float m = red[0]; __syncthreads();

  float lsum = 0.0f;
  #pragma unroll
  for (int j = 0; j < 8; ++j) { vals[j] = __expf(vals[j] - m); lsum += vals[j]; }
  red[t] = lsum; __syncthreads();
  for (int off = 128; off > 0; off >>= 1) {
    if (t < off) red[t] += red[t + off];
    __syncthreads();
  }
  const float inv = 1.0f / red[0];
  #pragma unroll
  for (int j = 0; j < 8; ++j)
    out[b * Sdim + t + j * 256] = vals[j] * inv;
}

// ---------------------------------------------------------------------------
extern "C" void kernel_launch(void* const* d_in, const int* in_sizes, int n_in,
                              void* d_out, int out_size, void* d_ws, size_t ws_size,
                              hipStream_t stream) {
  const float* hidden = (const float*)d_in[0];   // [1,32,1024]
  const float* enc    = (const float*)d_in[1];   // [2048,32,2048]
  const float* W      = (const float*)d_in[2];   // [1024,2048]
  // d_in[3] (bias) intentionally unused: adds a per-row constant to the
  // logits, which softmax cancels exactly.

  float* v      = (float*)d_ws;                  // 32*2048 f32 = 256 KB
  float* scores = v + Bdim * K2;                 // 32*2048 f32 = 256 KB

  proj_wmma   <<<32,                256, 0, stream>>>(hidden, W, v);
  scores_kernel<<<(Sdim * Bdim) / 8, 256, 0, stream>>>(enc, v, scores);
  softmax_kernel<<<Bdim,            256, 0, stream>>>(scores, (float*)d_out);
}